// SliceProjection_87050397156061
// MI455X (gfx1250) — compile-verified
//
#include <hip/hip_runtime.h>

typedef __attribute__((ext_vector_type(2))) float v2f;
typedef __attribute__((ext_vector_type(8))) float v8f;

// One wave32 computes a tile of 16 output pixels.
// Per chunk of 4 K-values:
//   A (16x4 f32, 2 VGPRs): lane l holds W[l&15, k0+2*(l>>4)] , W[..+1]  (contiguous float2)
//   B (4x16 f32, 2 VGPRs): lane l holds x[idx[l&15, k0+2*(l>>4)]], x[idx[..+1]]
//   C += A x B  -> diag(C)[i] accumulates sum_k W[i,k]*x[idx[i,k]]
// Diagonal of the 16x16 f32 accumulator (8 VGPRs):
//   row r (r<8)  -> lane r,    component r
//   row 8+r      -> lane 24+r, component r
__global__ __launch_bounds__(256) void slice_proj_wmma_kernel(
    const float* __restrict__ x,
    const float* __restrict__ weights,
    const int*   __restrict__ src_idx,
    float*       __restrict__ out,
    int K, int n_out)
{
    const int lane = threadIdx.x & 31;
    const int wave = threadIdx.x >> 5;          // 8 waves per block
    const int tile = blockIdx.x * 8 + wave;     // 16 pixels per tile
    const int pb   = tile * 16;
    if (pb >= n_out) return;                    // uniform per wave -> EXEC stays full

    const int row = lane & 15;                  // A-row / B-col owned by this lane
    const int hi  = lane >> 4;                  // which K-pair of the chunk
    const long long gp = (long long)pb + row;   // global (b*P + p) pixel index
    const float* __restrict__ wrow = weights + gp * (long long)K;
    const int*   __restrict__ irow = src_idx + gp * (long long)K;

    v8f c = {};                                 // 16x16 f32 accumulator
    const int K4 = K & ~3;
    for (int k0 = 0; k0 < K4; k0 += 4) {
        const int kk = k0 + 2 * hi;
        // streaming prefetch ~4 chunks ahead (covers the whole cacheline)
        __builtin_prefetch(wrow + kk + 16, 0, 0);
        __builtin_prefetch(irow + kk + 16, 0, 0);

        v2f a;
        a.x = wrow[kk];
        a.y = wrow[kk + 1];

        const int i0 = irow[kk];
        const int i1 = irow[kk + 1];
        v2f b;
        b.x = x[i0];
        b.y = x[i1];

        // D = A(16x4) * B(4x16) + C ; 8 args: neg_a, A, neg_b, B, c_mod, C, reuse_a, reuse_b
        c = __builtin_amdgcn_wmma_f32_16x16x4_f32(
                false, a, false, b, (short)0, c, false, false);
    }

    // Extract this lane's diagonal element (compile-time unrolled select).
    const int r = lane & 7;
    float diag = c[0];
#pragma unroll
    for (int j = 1; j < 8; ++j)
        if (r == j) diag = c[j];

    // Lanes 0..7 own rows 0..7 ; lanes 24..31 own rows 8..15.
    const bool writer = (lane < 8) || (lane >= 24);
    if (writer) {
        const int orow = (lane < 8) ? lane : (lane - 16);   // 0..15
        const long long g = (long long)pb + orow;
        const float* __restrict__ wr = weights + g * (long long)K;
        const int*   __restrict__ ir = src_idx + g * (long long)K;
        float tail = 0.0f;
        for (int k = K4; k < K; ++k)
            tail += wr[k] * x[ir[k]];
        out[pb + orow] = diag + tail;
    }
}

extern "C" void kernel_launch(void* const* d_in, const int* in_sizes, int n_in,
                              void* d_out, int out_size, void* d_ws, size_t ws_size,
                              hipStream_t stream) {
    const float* x       = (const float*)d_in[0];   // (256,256,256) f32
    const float* weights = (const float*)d_in[1];   // (B, P, K) f32
    const int*   src_idx = (const int*)d_in[2];     // (B, P, K) i32
    float*       out     = (float*)d_out;           // (B, P) f32

    const int K = (out_size > 0) ? (in_sizes[1] / out_size) : 0;

    const int tiles  = (out_size + 15) / 16;   // 16 pixels per wave
    const int blocks = (tiles + 7) / 8;        // 8 waves (256 threads) per block

    slice_proj_wmma_kernel<<<blocks, 256, 0, stream>>>(
        x, weights, src_idx, out, K, out_size);
}